// LearnableDiffusionLayer_1176821039618
// MI455X (gfx1250) — compile-verified
//
#include <hip/hip_runtime.h>
#include <hip/hip_bf16.h>

typedef float v2f __attribute__((ext_vector_type(2)));
typedef float v8f __attribute__((ext_vector_type(8)));

#define D 16

// ---------------------------------------------------------------------------
// Kernel 1: per 16x16 node tile, compute xw = X * diag(weight) with
// V_WMMA_F32_16X16X4_F32 (4 chained K=4 steps => K=16), and initialize the
// accumulator acc = x * (1 + self_loop_weight).
// One wave per tile; all branches (tile guard, full-tile epilogue) are
// wave-uniform so EXEC stays all-1 around the WMMAs and stores.
// ---------------------------------------------------------------------------
__global__ void __launch_bounds__(256)
node_precompute_kernel(const float* __restrict__ x,
                       const float* __restrict__ weight,
                       const float* __restrict__ slw_ptr,
                       float* __restrict__ xw,
                       float* __restrict__ acc,
                       int n_tiles, int n_nodes) {
    const int gid  = blockIdx.x * blockDim.x + threadIdx.x;
    const int wave = gid >> 5;
    if (wave >= n_tiles) return;               // wave-uniform branch

    const int lane = threadIdx.x & 31;
    const int n    = lane & 15;                // N (column) index, also A's M
    const int hi   = lane >> 4;                // half-wave selector
    const int base = wave << 4;                // first node row of this tile

    const float slw1 = 1.0f + *slw_ptr;
    const float wn   = weight[n];              // single unconditional load

    v8f c = {};
#pragma unroll
    for (int kb = 0; kb < 4; ++kb) {
        v2f a, b;
#pragma unroll
        for (int v = 0; v < 2; ++v) {
            // ISA 32-bit A 16x4 layout: lanes 0-15 hold K=v, lanes 16-31 hold K=v+2
            const int k   = kb * 4 + hi * 2 + v;     // global K in [0,16)
            const int row = base + n;                // A-matrix M = lane&15
            const int r   = row < n_nodes ? row : n_nodes - 1;
            a[v] = x[r * D + k];
            // B = diag(weight) slice: VALU select only (v_cndmask), no load
            b[v] = (k == n) ? wn : 0.0f;
        }
        c = __builtin_amdgcn_wmma_f32_16x16x4_f32(
                /*neg_a=*/false, a, /*neg_b=*/false, b,
                /*c_mod=*/(short)0, c, /*reuse_a=*/false, /*reuse_b=*/false);
    }

    if (base + 16 <= n_nodes) {
        // Full tile (always true for N % 16 == 0): no per-element guards.
        // xw stores follow the C/D layout: VGPR v -> M = v + hi*8, N = lane&15.
#pragma unroll
        for (int v = 0; v < 8; ++v) {
            const int row = base + v + hi * 8;
            xw[row * D + n] = c[v];
        }
        // acc init: the tile is 256 contiguous floats; do it coalesced as
        // 2 float4 load/store pairs per lane (GLOBAL_LOAD/STORE_B128).
        const float4* xs = (const float4*)(x + base * D);
        float4*       as = (float4*)(acc + base * D);
#pragma unroll
        for (int t = 0; t < 2; ++t) {
            const int idx = t * 32 + lane;
            float4 v4 = xs[idx];
            v4.x *= slw1; v4.y *= slw1; v4.z *= slw1; v4.w *= slw1;
            as[idx] = v4;
        }
    } else {
        // Tail tile fallback (never taken for N = 100000).
#pragma unroll
        for (int v = 0; v < 8; ++v) {
            const int row = base + v + hi * 8;
            if (row < n_nodes) {
                xw[row * D + n]  = c[v];
                acc[row * D + n] = x[row * D + n] * slw1;
            }
        }
    }
}

// ---------------------------------------------------------------------------
// Kernel 2: one thread per edge. Edge arrays streamed non-temporally (they
// are read exactly once: ~64 MB total, the only real HBM traffic). xw row is
// a 64B contiguous chunk (GLOBAL_LOAD_B128 x4, L2-resident 6.4 MB table).
// 16 relaxed agent-scope global_atomic_add_f32 into the L2-resident
// accumulator.
// ---------------------------------------------------------------------------
__global__ void __launch_bounds__(256)
edge_scatter_kernel(const long long* __restrict__ ei,
                    const float* __restrict__ probs,
                    const float* __restrict__ xw,
                    float* __restrict__ acc,
                    int n_edges) {
    const int e = blockIdx.x * blockDim.x + threadIdx.x;
    if (e >= n_edges) return;

    const long long s = __builtin_nontemporal_load(&ei[e]);
    const long long d = __builtin_nontemporal_load(&ei[(long long)n_edges + e]);
    const float     p = __builtin_nontemporal_load(&probs[e]);

    const float4* xr = (const float4*)(xw + s * D);
    const float4 v0 = xr[0];
    const float4 v1 = xr[1];
    const float4 v2 = xr[2];
    const float4 v3 = xr[3];

    float vals[D] = { v0.x * p, v0.y * p, v0.z * p, v0.w * p,
                      v1.x * p, v1.y * p, v1.z * p, v1.w * p,
                      v2.x * p, v2.y * p, v2.z * p, v2.w * p,
                      v3.x * p, v3.y * p, v3.z * p, v3.w * p };

    float* ao = acc + d * D;
#pragma unroll
    for (int i = 0; i < D; ++i) {
        __hip_atomic_fetch_add(ao + i, vals[i],
                               __ATOMIC_RELAXED, __HIP_MEMORY_SCOPE_AGENT);
    }
}

// ---------------------------------------------------------------------------
// Kernel 3: clamp accumulator to [0,1] in place (float4 granularity).
// ---------------------------------------------------------------------------
__global__ void __launch_bounds__(256)
clip_kernel(float* __restrict__ acc, int n4) {
    const int i = blockIdx.x * blockDim.x + threadIdx.x;
    if (i >= n4) return;
    float4 v = ((float4*)acc)[i];
    v.x = fminf(fmaxf(v.x, 0.0f), 1.0f);
    v.y = fminf(fmaxf(v.y, 0.0f), 1.0f);
    v.z = fminf(fmaxf(v.z, 0.0f), 1.0f);
    v.w = fminf(fmaxf(v.w, 0.0f), 1.0f);
    ((float4*)acc)[i] = v;
}

extern "C" void kernel_launch(void* const* d_in, const int* in_sizes, int n_in,
                              void* d_out, int out_size, void* d_ws, size_t ws_size,
                              hipStream_t stream) {
    const float*     x      = (const float*)d_in[0];
    const long long* ei     = (const long long*)d_in[1];  // [2, E] int64
    const float*     probs  = (const float*)d_in[2];
    const float*     weight = (const float*)d_in[3];
    const float*     slw    = (const float*)d_in[4];      // scalar on device
    float*           out    = (float*)d_out;              // [N, 16] accumulator
    float*           xw     = (float*)d_ws;               // [N, 16] scratch

    const int n_nodes = in_sizes[0] / D;        // 100000
    const int n_edges = in_sizes[1] / 2;        // 3200000
    const int n_tiles = (n_nodes + 15) / 16;    // 6250 waves

    // 8 waves (tiles) per 256-thread block
    const int grdA = (n_tiles + 7) / 8;
    node_precompute_kernel<<<grdA, 256, 0, stream>>>(x, weight, slw, xw, out,
                                                     n_tiles, n_nodes);

    const int grdB = (n_edges + 255) / 256;
    edge_scatter_kernel<<<grdB, 256, 0, stream>>>(ei, probs, xw, out, n_edges);

    const int n4   = (n_nodes * D) / 4;
    const int grdC = (n4 + 255) / 256;
    clip_kernel<<<grdC, 256, 0, stream>>>(out, n4);
}